// VisualGraphEmbedding_60129542661
// MI455X (gfx1250) — compile-verified
//
#include <hip/hip_runtime.h>
#include <math.h>

#define D 128
#define NGRAPH 32
#define IMGDIM 4096

typedef float v2f __attribute__((ext_vector_type(2)));
typedef float v8f __attribute__((ext_vector_type(8)));
typedef int   v4i __attribute__((ext_vector_type(4)));

#if defined(__has_builtin)
#if __has_builtin(__builtin_amdgcn_global_load_async_to_lds_b128) && \
    __has_builtin(__builtin_amdgcn_s_wait_asynccnt)
#define USE_ASYNC_LDS 1
#endif
#endif

// ---------------------------------------------------------------- fill
__global__ void fill_f32(float* __restrict__ p, float v, size_t n) {
  size_t i = (size_t)blockIdx.x * blockDim.x + threadIdx.x;
  size_t stride = (size_t)gridDim.x * blockDim.x;
  for (; i < n; i += stride) p[i] = v;
}

// ---------------------------------------------------------------- degree
__global__ void deg_accum(const int* __restrict__ dst, const float* __restrict__ ew,
                          float* __restrict__ deg, int E) {
  int i = blockIdx.x * blockDim.x + threadIdx.x;
  int stride = gridDim.x * blockDim.x;
  for (; i < E; i += stride) atomicAdd(&deg[dst[i]], ew[i]);
}

__global__ void deg_to_dis(float* __restrict__ deg, int N) {
  int i = blockIdx.x * blockDim.x + threadIdx.x;
  if (i < N) {
    float d = deg[i];
    deg[i] = d > 0.0f ? rsqrtf(fmaxf(d, 1e-12f)) : 0.0f;
  }
}

// ---------------------------------------------------------------- WMMA GEMM, M == 32
// Each wave owns one 16-col N-tile and BOTH 16-row M-tiles: every B fragment
// (2 global loads) feeds two V_WMMA_F32_16X16X4_F32 -> B (the big operand,
// e.g. 64MB W_img) is streamed from HBM exactly once.
__global__ __launch_bounds__(128) void wmma_gemm_m32(
    const float* __restrict__ A, const float* __restrict__ B,
    const float* __restrict__ bias, float* __restrict__ C, int N, int K) {
  const int wave = threadIdx.x >> 5;
  const int lane = threadIdx.x & 31;
  const int tn = blockIdx.x * 4 + wave;
  if (tn >= (N >> 4)) return;                   // wave-uniform: EXEC stays all-ones

  const int halfsel = lane >> 4;                // 0: lanes 0-15, 1: lanes 16-31
  const int sub = lane & 15;

  const float* Ap0 = A + (size_t)sub * K + (halfsel << 1);        // rows 0..15
  const float* Ap1 = A + (size_t)(16 + sub) * K + (halfsel << 1); // rows 16..31
  const float* Bp = B + (size_t)(halfsel << 1) * N + tn * 16 + sub;

  v8f c0 = {}, c1 = {};
  for (int k = 0; k < K; k += 4) {
    v2f b;
    b.x = Bp[0];
    b.y = Bp[N];
    v2f a0 = *(const v2f*)(Ap0 + k);
    v2f a1 = *(const v2f*)(Ap1 + k);
    c0 = __builtin_amdgcn_wmma_f32_16x16x4_f32(false, a0, false, b, (short)0, c0, false, false);
    c1 = __builtin_amdgcn_wmma_f32_16x16x4_f32(false, a1, false, b, (short)0, c1, false, false);
    Bp += (size_t)4 * N;
  }

  const int col = tn * 16 + sub;
  const int r0 = halfsel << 3;
  const float bv = bias ? bias[col] : 0.0f;
#pragma unroll
  for (int i = 0; i < 8; ++i) {
    C[(size_t)(r0 + i) * N + col] = c0[i] + bv;          // M-tile 0: rows 0..15
    C[(size_t)(16 + r0 + i) * N + col] = c1[i] + bv;     // M-tile 1: rows 16..31
  }
}

// ---------------------------------------------------------------- WMMA GEMM, N = K = 128
// GCN feature transform: xw = x @ W (40000x128 @ 128x128).
// Block = 4 waves = 64 rows x all 128 cols. W (64KB) is staged into LDS once
// per block via async global->LDS DMA (ASYNCcnt path); each wave then runs
// 8 WMMAs (one per N-tile) per single 8-byte global A-fragment load.
__global__ __launch_bounds__(128) void wmma_gcn_gemm(
    const float* __restrict__ A, const int* __restrict__ idxA,
    const float* __restrict__ B, float* __restrict__ C, int M) {
  __shared__ float sB[D * D];                   // 64 KB of the 320 KB/WGP LDS
  const int tid = threadIdx.x;

#ifdef USE_ASYNC_LDS
  for (int i = tid * 4; i < D * D; i += 128 * 4) {
    __builtin_amdgcn_global_load_async_to_lds_b128(
        (v4i*)(B + i), (v4i*)(sB + i), 0, 0);
  }
  __builtin_amdgcn_s_wait_asynccnt(0);
  __syncthreads();
#else
  for (int i = tid; i < (D * D) / 4; i += 128)
    ((float4*)sB)[i] = ((const float4*)B)[i];
  __syncthreads();
#endif

  const int wave = tid >> 5;
  const int lane = tid & 31;
  const int halfsel = lane >> 4;
  const int sub = lane & 15;

  int arow = blockIdx.x * 64 + wave * 16 + sub;
  if (arow > M - 1) arow = M - 1;               // clamp (reads only; stores guarded)
  if (idxA) arow = idxA[arow];                  // fused embedding gather (layer 1)
  const float* Ap = A + (size_t)arow * D + (halfsel << 1);

  v8f c[8] = {};
  for (int k = 0; k < D; k += 4) {
    v2f a = *(const v2f*)(Ap + k);              // one b64 global load per 8 WMMAs
    const int kb = k + (halfsel << 1);
#pragma unroll
    for (int t = 0; t < 8; ++t) {
      v2f b;
      b.x = sB[kb * D + t * 16 + sub];
      b.y = sB[(kb + 1) * D + t * 16 + sub];
      c[t] = __builtin_amdgcn_wmma_f32_16x16x4_f32(false, a, false, b, (short)0, c[t], false, false);
    }
  }

  const int rbase = blockIdx.x * 64 + wave * 16 + (halfsel << 3);
#pragma unroll
  for (int t = 0; t < 8; ++t) {
    const int col = t * 16 + sub;
#pragma unroll
    for (int i = 0; i < 8; ++i) {
      int r = rbase + i;
      if (r < M) C[(size_t)r * D + col] = c[t][i];
    }
  }
}

// ---------------------------------------------------------------- edge scatter
// acc[dst] += xw[src] * (dis[src]*ew*dis[dst]); one block (=128 lanes = D) per
// edge. xw/acc are ~20MB each -> L2-resident on the 192MB L2.
__global__ void scatter_edges(const float* __restrict__ xw, const int* __restrict__ src,
                              const int* __restrict__ dst, const float* __restrict__ ew,
                              const float* __restrict__ dis, float* __restrict__ acc, int E) {
  const int e = blockIdx.x;
  if (e >= E) return;
  const int f = threadIdx.x;
  const int s = src[e];
  const int d = dst[e];
  const float nrm = dis[s] * ew[e] * dis[d];
  atomicAdd(&acc[(size_t)d * D + f], xw[(size_t)s * D + f] * nrm);
}

// ---------------------------------------------------------------- GCN epilogue
// out = (relu?)(acc + xw*dis^2 + bias)   -- xw*dis^2 is the self-loop term
__global__ void gcn_finalize(const float* __restrict__ acc, const float* __restrict__ xw,
                             const float* __restrict__ dis, const float* __restrict__ bias,
                             float* __restrict__ out, int N, int relu) {
  const int i = blockIdx.x;
  const int f = threadIdx.x;
  const float ds = dis[i];
  float v = acc[(size_t)i * D + f] + xw[(size_t)i * D + f] * ds * ds + bias[f];
  if (relu) v = fmaxf(v, 0.0f);
  out[(size_t)i * D + f] = v;
}

// ---------------------------------------------------------------- mean pool
__global__ void pool_accum(const float* __restrict__ x, const int* __restrict__ batch,
                           float* __restrict__ sums, float* __restrict__ cnts, int N) {
  const int i = blockIdx.x;
  if (i >= N) return;
  const int f = threadIdx.x;
  const int b = batch[i];
  atomicAdd(&sums[(size_t)b * D + f], x[(size_t)i * D + f]);
  if (f == 0) atomicAdd(&cnts[b], 1.0f);
}

__global__ void pool_div(const float* __restrict__ sums, const float* __restrict__ cnts,
                         float* __restrict__ pooled) {
  const int b = blockIdx.x;
  const int f = threadIdx.x;
  pooled[(size_t)b * D + f] = sums[(size_t)b * D + f] / fmaxf(cnts[b], 1.0f);
}

// ---------------------------------------------------------------- row L2 normalize
__global__ void l2norm_rows(const float* __restrict__ in, float* __restrict__ out) {
  __shared__ float red[D];
  const int r = blockIdx.x;
  const int f = threadIdx.x;
  const float v = in[(size_t)r * D + f];
  red[f] = v * v;
  __syncthreads();
  for (int s = D / 2; s > 0; s >>= 1) {
    if (f < s) red[f] += red[f + s];
    __syncthreads();
  }
  out[(size_t)r * D + f] = v * rsqrtf(red[0]);
}

// ================================================================ host
extern "C" void kernel_launch(void* const* d_in, const int* in_sizes, int n_in,
                              void* d_out, int out_size, void* d_ws, size_t ws_size,
                              hipStream_t stream) {
  (void)n_in; (void)out_size; (void)ws_size;
  const float* images   = (const float*)d_in[0];
  const int*   node_ids = (const int*)  d_in[1];
  const int*   src      = (const int*)  d_in[2];
  const int*   dst      = (const int*)  d_in[3];
  const float* ew       = (const float*)d_in[4];
  const int*   batch    = (const int*)  d_in[5];
  const float* node_emb = (const float*)d_in[6];
  const float* W_img    = (const float*)d_in[7];
  const float* b_img    = (const float*)d_in[8];
  const float* W_i      = (const float*)d_in[9];
  const float* b_i      = (const float*)d_in[10];
  const float* W1 = (const float*)d_in[11]; const float* b1 = (const float*)d_in[12];
  const float* W2 = (const float*)d_in[13]; const float* b2 = (const float*)d_in[14];
  const float* W3 = (const float*)d_in[15]; const float* b3 = (const float*)d_in[16];
  const float* W_g = (const float*)d_in[17]; const float* b_g = (const float*)d_in[18];

  const int N = in_sizes[1];   // 40000 nodes
  const int E = in_sizes[2];   // 640000 edges
  float* out = (float*)d_out;  // [32,128] images ++ [32,128] graphs

  // ---- workspace carve-up (256B aligned) ----
  char* w = (char*)d_ws;
  size_t off = 0;
  auto alloc = [&](size_t bytes) -> float* {
    float* p = (float*)(w + off);
    off += (bytes + 255) & ~(size_t)255;
    return p;
  };
  float* q      = alloc((size_t)NGRAPH * IMGDIM * sizeof(float)); // 512 KB
  float* xbuf   = alloc((size_t)N * D * sizeof(float));           // 20.5 MB
  float* xw     = alloc((size_t)N * D * sizeof(float));           // 20.5 MB
  float* acc    = alloc((size_t)N * D * sizeof(float));           // 20.5 MB
  float* dis    = alloc((size_t)N * sizeof(float));
  float* sums   = alloc((size_t)NGRAPH * D * sizeof(float));
  float* cnts   = alloc((size_t)NGRAPH * sizeof(float));
  float* pooled = alloc((size_t)NGRAPH * D * sizeof(float));
  float* rimg   = alloc((size_t)NGRAPH * D * sizeof(float));
  float* gbuf   = alloc((size_t)NGRAPH * D * sizeof(float));

  // ================= image branch =================
  {
    // q = images @ W_img + b_img   (32 x 4096, K = 4096); W_img read once.
    wmma_gemm_m32<<<(IMGDIM / 16 + 3) / 4, 128, 0, stream>>>(
        images, W_img, b_img, q, IMGDIM, IMGDIM);
    // rimg = q @ W_i + b_i         (32 x 128, K = 4096)
    wmma_gemm_m32<<<(D / 16 + 3) / 4, 128, 0, stream>>>(
        q, W_i, b_i, rimg, D, IMGDIM);
    l2norm_rows<<<NGRAPH, D, 0, stream>>>(rimg, out);
  }

  // ================= graph branch =================
  // degree / symmetric normalization (invariant across the 3 layers)
  fill_f32<<<256, 256, 0, stream>>>(dis, 1.0f, (size_t)N);           // self-loop weight
  deg_accum<<<2048, 256, 0, stream>>>(dst, ew, dis, E);
  deg_to_dis<<<(N + 255) / 256, 256, 0, stream>>>(dis, N);

  const float* Ws[3] = {W1, W2, W3};
  const float* bs[3] = {b1, b2, b3};
  const int gcnBlocks = (N + 63) / 64;                               // 625
  for (int l = 0; l < 3; ++l) {
    const float* Ain = (l == 0) ? node_emb : xbuf;
    const int* idx   = (l == 0) ? node_ids : nullptr;                // fused embedding gather
    wmma_gcn_gemm<<<gcnBlocks, 128, 0, stream>>>(Ain, idx, Ws[l], xw, N);
    fill_f32<<<4096, 256, 0, stream>>>(acc, 0.0f, (size_t)N * D);
    scatter_edges<<<E, D, 0, stream>>>(xw, src, dst, ew, dis, acc, E);
    gcn_finalize<<<N, D, 0, stream>>>(acc, xw, dis, bs[l], xbuf, N, l < 2 ? 1 : 0);
  }

  // mean pool + projection + L2 normalize
  fill_f32<<<16, 256, 0, stream>>>(sums, 0.0f, (size_t)NGRAPH * D);
  fill_f32<<<1, 32, 0, stream>>>(cnts, 0.0f, (size_t)NGRAPH);
  pool_accum<<<N, D, 0, stream>>>(xbuf, batch, sums, cnts, N);
  pool_div<<<NGRAPH, D, 0, stream>>>(sums, cnts, pooled);
  wmma_gemm_m32<<<(D / 16 + 3) / 4, 128, 0, stream>>>(
      pooled, W_g, b_g, gbuf, D, D);
  l2norm_rows<<<NGRAPH, D, 0, stream>>>(gbuf, out + (size_t)NGRAPH * D);
}